// ContrastiveLoss_45406394253765
// MI455X (gfx1250) — compile-verified
//
#include <hip/hip_runtime.h>
#include <math.h>

// ---------------------------------------------------------------------------
// ContrastiveLoss on MI455X (gfx1250, wave32).
// Bandwidth-bound (~292 MB @ 23.3 TB/s => ~12.5us floor). Uses
// V_WMMA_F32_16X16X4_F32 Gram accumulation: tile = [src; 15 passage rows],
// C += A*A^T over K. D[0][j] = src.row_j dots, D[j][j] = row norms^2,
// D[0][0] = ||src||^2. For this shape the A and B fragment layouts coincide,
// so the same v2f register feeds SRC0 and SRC1.
// ---------------------------------------------------------------------------

typedef __attribute__((ext_vector_type(2))) float v2f;
typedef __attribute__((ext_vector_type(8))) float v8f;

#define ALPHA_F   0.8f
#define B_SZ      512
#define D_SZ      1024
#define P_SZ      16
#define N_SZ      128
#define NROWS     145   // 16 rel + 128 irr + 1 tgt
#define TILES     10    // ceil(145 / 15)
#define NTHREADS  320   // 10 waves, one wave per tile

__device__ __forceinline__ float v8_get(v8f v, int i) {
    float x = v[0];
    x = (i == 1) ? v[1] : x;
    x = (i == 2) ? v[2] : x;
    x = (i == 3) ? v[3] : x;
    x = (i == 4) ? v[4] : x;
    x = (i == 5) ? v[5] : x;
    x = (i == 6) ? v[6] : x;
    x = (i == 7) ? v[7] : x;
    return x;
}

__global__ __launch_bounds__(NTHREADS)
void contrastive_gram_kernel(const float* __restrict__ src,
                             const float* __restrict__ tgt,
                             const float* __restrict__ rel,
                             const float* __restrict__ irr,
                             float* __restrict__ per_b_loss) {
    __shared__ float s_dot[TILES][16];   // D[0][j]  : src . row_j
    __shared__ float s_sq [TILES][16];   // D[j][j]  : ||row_j||^2 (slot0 = ||src||^2)
    __shared__ float s_pos[TILES];
    __shared__ float s_neg[TILES];
    __shared__ float s_diag;

    const int b    = blockIdx.x;
    const int tid  = threadIdx.x;
    const int wave = tid >> 5;          // tile index 0..9
    const int lane = tid & 31;
    const int slot = lane & 15;         // row slot within tile (0 = src)
    const int half = lane >> 4;         // 0: K={0,1}, 1: K={2,3}

    // --- per-lane row pointer ------------------------------------------------
    const float* rowp;
    if (slot == 0) {
        rowp = src + (size_t)b * D_SZ;
    } else {
        int g = wave * 15 + slot - 1;   // global row id 0..144
        if (g > 144) g = 144;           // pad rows duplicate tgt (outputs ignored)
        if (g < 16)        rowp = rel + ((size_t)b * P_SZ + g)        * D_SZ;
        else if (g < 144)  rowp = irr + ((size_t)b * N_SZ + (g - 16)) * D_SZ;
        else               rowp = tgt + (size_t)b * D_SZ;
    }
    rowp += 2 * half;                   // A-frag K offset for this half-wave

    // --- Gram accumulation over K: C += A_chunk * A_chunk^T ------------------
    v8f c = {0.f, 0.f, 0.f, 0.f, 0.f, 0.f, 0.f, 0.f};
    #pragma unroll 8
    for (int k = 0; k < D_SZ; k += 4) {
        v2f a = *(const v2f*)(rowp + k);
        // (neg_a, A, neg_b, B, c_mod, C, reuse_a, reuse_b); A-frag == B-frag here
        c = __builtin_amdgcn_wmma_f32_16x16x4_f32(
                false, a, false, a, (short)0, c, false, false);
    }

    // --- Extract D[0][j] and D[j][j] per documented C/D VGPR layout ----------
    // lanes 0-15: N=lane,   VGPR r -> M=r
    // lanes16-31: N=lane-16,VGPR r -> M=8+r
    if (lane < 16) {
        s_dot[wave][lane] = c[0];                       // M=0 row
        if (lane < 8) s_sq[wave][lane] = v8_get(c, lane);       // D[n][n], n<8
    } else {
        int n = lane - 16;
        if (n >= 8) s_sq[wave][n] = v8_get(c, n - 8);           // D[n][n], n>=8
    }
    __syncthreads();

    // --- Finalize: cosine sims, exp sums, loss -------------------------------
    float pos = 0.f, neg = 0.f;
    const float inv_snorm = rsqrtf(fmaxf(s_sq[0][0], 1e-24f));  // 1/||src||
    if (tid < TILES * 16) {
        int t  = tid >> 4;
        int sl = tid & 15;
        if (sl > 0) {
            int g = t * 15 + sl - 1;
            if (g <= 144) {
                float sim = s_dot[t][sl] * inv_snorm *
                            rsqrtf(fmaxf(s_sq[t][sl], 1e-24f));
                if (g < 16)        pos = expf(sim);
                else if (g < 144)  neg = expf(sim);
                else               s_diag = sim;        // src.tgt cosine (one thread)
            }
        }
    }
    // deterministic wave32 reduction
    #pragma unroll
    for (int m = 16; m > 0; m >>= 1) {
        pos += __shfl_xor(pos, m, 32);
        neg += __shfl_xor(neg, m, 32);
    }
    if (lane == 0) { s_pos[wave] = pos; s_neg[wave] = neg; }
    __syncthreads();

    if (tid == 0) {
        float P = 0.f, Ng = 0.f;
        #pragma unroll
        for (int i = 0; i < TILES; ++i) { P += s_pos[i]; Ng += s_neg[i]; }
        float pos_score = 1.f + P;
        float loss_pos  = logf(pos_score);
        float loss_neg  = logf(pos_score + Ng);
        per_b_loss[b] = -(ALPHA_F * s_diag + (1.f - ALPHA_F) * (loss_pos - loss_neg));
    }
}

__global__ __launch_bounds__(256)
void mean_reduce_kernel(const float* __restrict__ per_b, float* __restrict__ out) {
    __shared__ float sm[256];
    int tid = threadIdx.x;
    sm[tid] = per_b[tid] + per_b[tid + 256];
    __syncthreads();
    #pragma unroll
    for (int s = 128; s > 0; s >>= 1) {
        if (tid < s) sm[tid] += sm[tid + s];
        __syncthreads();
    }
    if (tid == 0) out[0] = sm[0] * (1.f / (float)B_SZ);
}

extern "C" void kernel_launch(void* const* d_in, const int* in_sizes, int n_in,
                              void* d_out, int out_size, void* d_ws, size_t ws_size,
                              hipStream_t stream) {
    const float* src = (const float*)d_in[0];   // [B, D]
    const float* tgt = (const float*)d_in[1];   // [B, D]
    const float* rel = (const float*)d_in[2];   // [B, P, D]
    const float* irr = (const float*)d_in[3];   // [B, N, D]
    float* ws  = (float*)d_ws;                  // 512 per-b losses
    float* out = (float*)d_out;

    contrastive_gram_kernel<<<B_SZ, NTHREADS, 0, stream>>>(src, tgt, rel, irr, ws);
    mean_reduce_kernel<<<1, 256, 0, stream>>>(ws, out);
}